// Detect_42752104464942
// MI455X (gfx1250) — compile-verified
//
#include <hip/hip_runtime.h>
#include <stdint.h>

// SSD Detect post-processing for MI455X (gfx1250, wave32).
// One workgroup per (batch, class) pair. Score column DMA'd into LDS via the
// Tensor Data Mover (strided-column tile descriptor), falling back to async
// global->LDS loads. Exact top-K via 4-round radix select + bitonic sort.

#define BDIM 256
#define NWAVES (BDIM / 32)

constexpr int B_ = 8;
constexpr int P_ = 25575;
constexpr int C_ = 81;
constexpr int K_ = 200;
constexpr float THRESH = 0.01f;
constexpr float VAR0 = 0.1f;
constexpr float VAR1 = 0.2f;
constexpr int P_PAD = 25600;   // 102400 B of LDS for the score column
constexpr int SORT_N = 512;    // bitonic sort width (>= K + tie slack)

typedef __attribute__((address_space(1))) int gint_t;
typedef __attribute__((address_space(3))) int lint_t;
typedef unsigned int u32x4 __attribute__((ext_vector_type(4)));
typedef int i32x8 __attribute__((ext_vector_type(8)));
typedef int i32x4 __attribute__((ext_vector_type(4)));

#if __has_builtin(__builtin_amdgcn_tensor_load_to_lds)
#define USE_TDM 1
#endif

// Monotone unsigned key for float ordering (handles negatives for safety).
__device__ __forceinline__ unsigned fkey(float f) {
  unsigned u = __float_as_uint(f);
  return u ^ ((unsigned)((int)u >> 31) | 0x80000000u);
}

__device__ __forceinline__ void write_entry(float* __restrict__ out, int bc,
                                            unsigned k, float sc, unsigned p,
                                            const float* __restrict__ loc,
                                            const float* __restrict__ prior,
                                            int b) {
  const float4 l  = ((const float4*)loc)[(size_t)b * P_ + p];
  const float4 pr = ((const float4*)prior)[p];
  float cx = pr.x + l.x * VAR0 * pr.z;
  float cy = pr.y + l.y * VAR0 * pr.w;
  float w  = pr.z * __expf(l.z * VAR1);
  float h  = pr.w * __expf(l.w * VAR1);
  float xmin = cx - 0.5f * w;
  float ymin = cy - 0.5f * h;
  float* o = out + ((size_t)bc * K_ + k) * 5;
  o[0] = sc;
  o[1] = xmin;
  o[2] = ymin;
  o[3] = xmin + w;
  o[4] = ymin + h;
}

__global__ __launch_bounds__(BDIM) void detect_topk_kernel(
    const float* __restrict__ loc, const float* __restrict__ conf,
    const float* __restrict__ prior, float* __restrict__ out) {
  __shared__ float s_scores[P_PAD];          // 102400 B: whole class column
  __shared__ unsigned s_hist[256];           // radix histogram
  __shared__ unsigned long long s_sort[SORT_N];  // (key<<32)|(~idx)
  __shared__ unsigned s_wscan[NWAVES];
  __shared__ unsigned s_cnt;
  __shared__ unsigned s_pivot;
  __shared__ unsigned s_remaining;
  __shared__ unsigned s_nabove;
  __shared__ unsigned s_base;

  const int tid = threadIdx.x;
  const int bc = blockIdx.x;          // 0 .. B*C-1
  const int b = bc / C_;
  const int c = bc % C_;

  // ---------------- Phase 0: DMA the score column conf[b, :, c] into LDS --
  const float* gcol = conf + (size_t)b * P_ * C_ + c;   // stride C floats

#if defined(USE_TDM)
  {
    // Each wave TDM-loads its 1/8 slice of the column as a [rows x 1] tile
    // of a row-major [P x C] tensor (tensor_dim0_stride = C).
    const int wid = __builtin_amdgcn_readfirstlane((int)(tid >> 5));
    const int CHUNK = (P_ + NWAVES - 1) / NWAVES;          // 3197
    const int row0 = wid * CHUNK;
    int rows = P_ - row0;
    if (rows > CHUNK) rows = CHUNK;

    unsigned long long ga =
        (unsigned long long)(uintptr_t)(gcol + (size_t)row0 * C_);
    unsigned lds_base = (unsigned)(size_t)(lint_t*)&s_scores[row0];

    // D# Group 0: count=1 | lds_addr | global_addr[56:0] | type=2
    u32x4 g0;
    g0[0] = 1u;                                            // count=1, user
    g0[1] = lds_base;                                      // lds_addr
    g0[2] = (unsigned)(ga & 0xFFFFFFFFull);                // global_addr lo
    g0[3] = ((unsigned)((ga >> 32) & 0x01FFFFFFull)) | (2u << 30);  // + type=2

    // D# Group 1: data_size=4B, tensor [C x rows], tile [1 x rows], stride0=C
    const unsigned td0 = (unsigned)C_;       // tensor_dim0 (x extent)
    const unsigned td1 = (unsigned)rows;     // tensor_dim1 (y extent)
    i32x8 g1;
    g1[0] = (int)(2u << 16);                               // data_size=2 (4B)
    g1[1] = (int)((td0 & 0xFFFFu) << 16);                  // abar=0 | dim0 lo16
    g1[2] = (int)((td0 >> 16) | ((td1 & 0xFFFFu) << 16));  // dim0 hi | dim1 lo
    g1[3] = (int)((td1 >> 16) | (1u << 16));               // dim1 hi | tile0=1
    g1[4] = (int)(td1 & 0xFFFFu);                          // tile1=rows|tile2=0
    g1[5] = (int)(unsigned)C_;                             // stride0 lo32
    g1[6] = 0;                                             // stride0 hi|s1 lo
    g1[7] = 0;                                             // stride1 hi
    i32x4 gz;
    gz[0] = 0; gz[1] = 0; gz[2] = 0; gz[3] = 0;
    i32x8 gz8;
    gz8[0] = 0; gz8[1] = 0; gz8[2] = 0; gz8[3] = 0;
    gz8[4] = 0; gz8[5] = 0; gz8[6] = 0; gz8[7] = 0;
    // clang-23 / therock toolchain: 6-arg form (g0, g1, g2, g3, g4, cpol)
    __builtin_amdgcn_tensor_load_to_lds(g0, g1, gz, gz, gz8, 0);
  }
#else
  for (int p = tid; p < P_; p += BDIM) {
#if __has_builtin(__builtin_amdgcn_global_load_async_to_lds_b32)
    __builtin_amdgcn_global_load_async_to_lds_b32(
        (gint_t*)(gcol + (size_t)p * C_),
        (lint_t*)&s_scores[p],
        /*offset=*/0, /*cpol=*/0);
#else
    s_scores[p] = gcol[(size_t)p * C_];
#endif
  }
#endif

  if (tid == 0) {
    s_cnt = 0u;
    s_nabove = 0u;
    s_base = 0u;
    s_pivot = 0u;
    s_remaining = (unsigned)K_;
  }

#if defined(USE_TDM)
#if __has_builtin(__builtin_amdgcn_s_wait_tensorcnt)
  __builtin_amdgcn_s_wait_tensorcnt(0);
#else
  asm volatile("s_wait_tensorcnt 0" ::: "memory");
#endif
#else
#if __has_builtin(__builtin_amdgcn_s_wait_asynccnt)
  __builtin_amdgcn_s_wait_asynccnt(0);
#else
  asm volatile("s_wait_asynccnt 0" ::: "memory");
#endif
#endif
  __syncthreads();

  // ---------------- Phase 1: count scores above threshold -----------------
  unsigned mycnt = 0;
  for (int p = tid; p < P_; p += BDIM)
    mycnt += (s_scores[p] > THRESH) ? 1u : 0u;
  atomicAdd(&s_cnt, mycnt);
  __syncthreads();
  const unsigned count = s_cnt;

  if (count <= (unsigned)K_) {
    // ========== Case A: stable stream compaction in prior order ==========
    const unsigned lane = tid & 31u;
    const unsigned wid = tid >> 5;
    for (int base = 0; base < P_; base += BDIM) {
      int p = base + tid;
      bool pass = (p < P_) && (s_scores[p] > THRESH);
      unsigned bmask = (unsigned)__ballot(pass);        // wave32
      unsigned prefix = __popc(bmask & ((1u << lane) - 1u));
      if (lane == 0) s_wscan[wid] = __popc(bmask);
      __syncthreads();
      if (tid == 0) {
        unsigned acc = s_base;
        for (int w = 0; w < NWAVES; ++w) {
          unsigned t = s_wscan[w];
          s_wscan[w] = acc;
          acc += t;
        }
        s_base = acc;
      }
      __syncthreads();
      if (pass) {
        unsigned pos = s_wscan[wid] + prefix;
        if (pos < (unsigned)K_)
          write_entry(out, bc, pos, s_scores[p], (unsigned)p, loc, prior, b);
      }
      __syncthreads();
    }
    // zero-fill unused slots [count, K)
    for (unsigned k = count + (unsigned)tid; k < (unsigned)K_; k += BDIM) {
      float* o = out + ((size_t)bc * K_ + k) * 5;
      o[0] = 0.f; o[1] = 0.f; o[2] = 0.f; o[3] = 0.f; o[4] = 0.f;
    }
  } else {
    // ========== Case B: exact top-K (sorted desc, ties by low index) =====
    // 4-round byte-wise radix select on monotone keys, entirely out of LDS.
    for (int round = 0; round < 4; ++round) {
      const int shift = 24 - 8 * round;
      for (int i = tid; i < 256; i += BDIM) s_hist[i] = 0u;
      __syncthreads();
      const unsigned pk = s_pivot;
      for (int p = tid; p < P_; p += BDIM) {
        unsigned key = fkey(s_scores[p]);
        if (round == 0 || (key >> (shift + 8)) == (pk >> (shift + 8)))
          atomicAdd(&s_hist[(key >> shift) & 0xFFu], 1u);
      }
      __syncthreads();
      if (tid == 0) {
        unsigned rem = s_remaining, cum = 0;
        int bin = 255;
        for (; bin > 0; --bin) {
          unsigned h = s_hist[bin];
          if (cum + h >= rem) break;
          cum += h;
        }
        s_pivot = pk | ((unsigned)bin << shift);
        s_remaining = rem - cum;
      }
      __syncthreads();
    }
    const unsigned kth = s_pivot;   // exact key of the K-th largest element

    // Collect everything >= kth (<= 199 strictly-greater + ties).
    for (int p = tid; p < P_; p += BDIM) {
      unsigned key = fkey(s_scores[p]);
      if (key >= kth) {
        unsigned pos = atomicAdd(&s_nabove, 1u);
        if (pos < (unsigned)SORT_N)
          s_sort[pos] = ((unsigned long long)key << 32) |
                        (unsigned long long)(0xFFFFFFFFu - (unsigned)p);
      }
    }
    __syncthreads();
    unsigned nf = s_nabove;
    if (nf > (unsigned)SORT_N) nf = (unsigned)SORT_N;
    for (int i = tid; i < SORT_N; i += BDIM)
      if ((unsigned)i >= nf) s_sort[i] = 0ull;   // sinks to the bottom

    // Bitonic sort, descending by (key, then ascending index).
    for (unsigned kk = 2; kk <= (unsigned)SORT_N; kk <<= 1) {
      for (unsigned j = kk >> 1; j > 0; j >>= 1) {
        __syncthreads();
        for (unsigned i = tid; i < (unsigned)SORT_N; i += BDIM) {
          unsigned ixj = i ^ j;
          if (ixj > i) {
            unsigned long long a = s_sort[i];
            unsigned long long bb = s_sort[ixj];
            bool desc = ((i & kk) == 0);
            if (desc ? (a < bb) : (a > bb)) {
              s_sort[i] = bb;
              s_sort[ixj] = a;
            }
          }
        }
      }
    }
    __syncthreads();

    for (int k = tid; k < K_; k += BDIM) {
      unsigned long long e = s_sort[k];
      unsigned p = 0xFFFFFFFFu - (unsigned)(e & 0xFFFFFFFFull);
      write_entry(out, bc, (unsigned)k, s_scores[p], p, loc, prior, b);
    }
  }
}

extern "C" void kernel_launch(void* const* d_in, const int* in_sizes, int n_in,
                              void* d_out, int out_size, void* d_ws,
                              size_t ws_size, hipStream_t stream) {
  (void)in_sizes; (void)n_in; (void)out_size; (void)d_ws; (void)ws_size;
  const float* loc   = (const float*)d_in[0];   // [8, 25575, 4]
  const float* conf  = (const float*)d_in[1];   // [8, 25575, 81]
  const float* prior = (const float*)d_in[2];   // [1, 25575, 4]
  float* out = (float*)d_out;                   // [8, 81, 200, 5]

  detect_topk_kernel<<<dim3(B_ * C_), dim3(BDIM), 0, stream>>>(loc, conf,
                                                               prior, out);
}